// RoIRingPool_76149770158213
// MI455X (gfx1250) — compile-verified
//
#include <hip/hip_runtime.h>
#include <cstdint>

#define PH 7
#define PW 7
#define CCH 256
#define HH 32
#define WW 32
#define NEGF (-3.4028234663852886e+38f)   // -FLT_MAX == jnp.finfo(f32).min

// One wave32 per (roi, channel). lane == x column (W == 32 == wave size).
// Block = 256 threads = 8 waves = 8 channels of one ROI.
// Grid  = (R, C/8).
__global__ __launch_bounds__(256) void roi_ring_pool_kernel(
    const float* __restrict__ feat,   // [B, 256, 32, 32]
    const float* __restrict__ rois,   // [R, 5]  (b, x1, y1, x2, y2)
    float* __restrict__ out,          // [R, 256, 7, 7]
    int R)
{
    __shared__ __align__(16) float planes[8 * HH * WW];  // 8 waves x 4KB plane
    __shared__ float res[8 * 64];                        // 49 results per wave

    const int r    = blockIdx.x;
    const int wv   = threadIdx.x >> 5;
    const int lane = threadIdx.x & 31;
    const int c    = blockIdx.y * 8 + wv;

    // ---------------- ROI parameters (uniform within wave) ----------------
    const float r0 = rois[r * 5 + 0], r1 = rois[r * 5 + 1], r2 = rois[r * 5 + 2],
                r3 = rois[r * 5 + 3], r4 = rois[r * 5 + 4];
    const int   b  = (int)r0;
    const float s  = 1.0f / 16.0f;                 // SPATIAL_SCALE
    const float SO = 1.8f, SI = 1.0f;              // SCALE_OUTER / SCALE_INNER
    const float cx = (r1 + r3) * 0.5f, cy = (r2 + r4) * 0.5f;
    const float wh = (r3 - r1) * 0.5f, hh = (r4 - r2) * 0.5f;
    // roundf == sign(x)*floor(|x|+0.5) (half away from zero), matches _round_half
    const float sw = roundf((cx - wh * SO) * s);
    const float sh = roundf((cy - hh * SO) * s);
    const float ew = roundf((cx + wh * SO) * s);
    const float eh = roundf((cy + hh * SO) * s);
    const int  isw = (int)roundf((cx - wh * SI) * s);
    const int  ish = (int)roundf((cy - hh * SI) * s);
    const int  iew = (int)roundf((cx + wh * SI) * s);
    const int  ieh = (int)roundf((cy + hh * SI) * s);
    const float roiW = fmaxf(ew - sw + 1.0f, 1.0f);
    const float roiH = fmaxf(eh - sh + 1.0f, 1.0f);
    const float binH = roiH / (float)PH;
    const float binW = roiW / (float)PW;

    int hs[PH], he[PH], wsx[PW], wex[PW];
    int cntH[PH], ovH[PH], cntW[PW], ovW[PW];
#pragma unroll
    for (int p = 0; p < PH; ++p) {
        float a = fminf(fmaxf(floorf((float)p * binH) + sh, 0.f), (float)HH);
        float e = fminf(fmaxf(ceilf((float)(p + 1) * binH) + sh, 0.f), (float)HH);
        hs[p] = (int)a; he[p] = (int)e;
        cntH[p] = max(he[p] - hs[p], 0);
        ovH[p]  = max(min(he[p], ieh) - max(hs[p], ish + 1), 0);
    }
#pragma unroll
    for (int q = 0; q < PW; ++q) {
        float a = fminf(fmaxf(floorf((float)q * binW) + sw, 0.f), (float)WW);
        float e = fminf(fmaxf(ceilf((float)(q + 1) * binW) + sw, 0.f), (float)WW);
        wsx[q] = (int)a; wex[q] = (int)e;
        cntW[q] = max(wex[q] - wsx[q], 0);
        ovW[q]  = max(min(wex[q], iew) - max(wsx[q], isw + 1), 0);
    }

    // ------------- async DMA of this wave's feature plane to LDS ----------
    // 4KB plane: 8 x (32 lanes x 16B) = 8 global_load_async_to_lds_b128.
    const float* gplane  = feat + (((size_t)b * CCH + c) * (HH * WW));
    unsigned     ldsBase = (unsigned)(uintptr_t)(&planes[wv * (HH * WW)]);
#pragma unroll
    for (int i = 0; i < 8; ++i) {
        unsigned     loff = ldsBase + (unsigned)(i * 512 + lane * 16);
        const float* g    = gplane + (i * 128 + lane * 4);
        asm volatile("global_load_async_to_lds_b128 %0, %1, off"
                     :: "v"(loff), "v"(g) : "memory");
    }
    asm volatile("s_wait_asynccnt 0x0" ::: "memory");

    // ------------- separable masked max: rows -> 7 row-bin accumulators ---
    const float* lp = &planes[wv * (HH * WW)];
    float rowmax[PH];
#pragma unroll
    for (int p = 0; p < PH; ++p) rowmax[p] = NEGF;

    const bool exw = (lane > isw) && (lane < iew);   // ring column exclusion
    const int y0 = hs[0], y1 = he[PH - 1];
    for (int y = y0; y < y1; ++y) {
        float v   = lp[y * WW + lane];               // ds_load_b32, conflict-free
        bool  exh = (y > ish) && (y < ieh);
        float vm  = (exh && exw) ? NEGF : v;
#pragma unroll
        for (int p = 0; p < PH; ++p)
            if (y >= hs[p] && y < he[p]) rowmax[p] = fmaxf(rowmax[p], vm);
    }

    // ------------- 49 masked cross-lane max reductions (wave32) -----------
#pragma unroll
    for (int p = 0; p < PH; ++p) {
#pragma unroll
        for (int q = 0; q < PW; ++q) {
            float t = (lane >= wsx[q] && lane < wex[q]) ? rowmax[p] : NEGF;
#pragma unroll
            for (int m = 16; m >= 1; m >>= 1)
                t = fmaxf(t, __shfl_xor(t, m, 32));
            // bin valid iff it contains at least one non-excluded pixel
            bool valid = (cntH[p] * cntW[q]) > (ovH[p] * ovW[q]);
            if (lane == 0) res[wv * 64 + p * PW + q] = valid ? t : 0.0f;
        }
    }

    // ------------- coalesced writeback of the contiguous 49-float block ---
    const size_t obase = ((size_t)r * CCH + c) * (PH * PW);
    for (int i = lane; i < PH * PW; i += 32)
        out[obase + i] = res[wv * 64 + i];
}

extern "C" void kernel_launch(void* const* d_in, const int* in_sizes, int n_in,
                              void* d_out, int out_size, void* d_ws, size_t ws_size,
                              hipStream_t stream) {
    const float* feat = (const float*)d_in[0];   // [B,256,32,32] f32
    const float* rois = (const float*)d_in[1];   // [R,5] f32
    float*       out  = (float*)d_out;           // [R,256,7,7] f32
    const int R = in_sizes[1] / 5;
    dim3 grid((unsigned)R, CCH / 8);
    roi_ring_pool_kernel<<<grid, 256, 0, stream>>>(feat, rois, out, R);
}